// MultiHeadAttentionLayer_47974784696414
// MI455X (gfx1250) — compile-verified
//
#include <hip/hip_runtime.h>

typedef __attribute__((ext_vector_type(2))) float v2f;
typedef __attribute__((ext_vector_type(8))) float v8f;

#define N_NODES 50000
#define N_EDGES 800000
#define DIM     128
#define EDIM    64
#define HEADS   4
#define HU      128   // heads * units
#define UOUT    32

// ---- monotonic uint encoding of float for atomic max over signed floats ----
__device__ __forceinline__ unsigned f32_order(float f) {
  unsigned u = __float_as_uint(f);
  return (u & 0x80000000u) ? ~u : (u | 0x80000000u);
}
__device__ __forceinline__ float f32_unorder(unsigned u) {
  return __uint_as_float((u & 0x80000000u) ? (u ^ 0x80000000u) : ~u);
}
#define ORDER_NEG_INF 0x007FFFFFu  // f32_order(-inf)

// ---------------------------------------------------------------------------
// 0) init accumulators
// ---------------------------------------------------------------------------
__global__ void init_ws(float* __restrict__ num, float* __restrict__ den,
                        unsigned* __restrict__ smaxu) {
  int i = blockIdx.x * blockDim.x + threadIdx.x;
  if (i < N_NODES * HU) num[i] = 0.0f;
  if (i < N_NODES * HEADS) { den[i] = 0.0f; smaxu[i] = ORDER_NEG_INF; }
}

// ---------------------------------------------------------------------------
// 1) Q/K/V projections: X[N,128] @ W[128,128] + b, fp32 WMMA 16x16x4
//    grid (N/16, 3), block 256 (8 waves = 8 column tiles of 16)
// ---------------------------------------------------------------------------
__global__ __launch_bounds__(256) void proj_qkv(
    const float* __restrict__ X,
    const float* __restrict__ Wq, const float* __restrict__ bq,
    const float* __restrict__ Wk, const float* __restrict__ bk,
    const float* __restrict__ Wv, const float* __restrict__ bv,
    float* __restrict__ Q, float* __restrict__ K, float* __restrict__ V) {
  __shared__ float xt[16 * DIM];  // 8KB node tile, shared by all 8 waves
  const int n0 = blockIdx.x * 16;
  const float* W; const float* bias; float* out;
  if (blockIdx.y == 0)      { W = Wq; bias = bq; out = Q; }
  else if (blockIdx.y == 1) { W = Wk; bias = bk; out = K; }
  else                      { W = Wv; bias = bv; out = V; }

  for (int i = threadIdx.x; i < 16 * DIM; i += 256) xt[i] = X[n0 * DIM + i];
  __syncthreads();

  const int wave = threadIdx.x >> 5, lane = threadIdx.x & 31;
  const int colg = wave * 16 + (lane & 15);   // output column
  const int row  = lane & 15;                 // A row (both lane halves)
  const int koff = (lane >> 4) * 2;           // K sub-offset per lane half

  v8f c;
  const float bb = bias[colg];
  #pragma unroll
  for (int j = 0; j < 8; ++j) c[j] = bb;      // bias folded into accumulator

  #pragma unroll
  for (int kk = 0; kk < DIM; kk += 4) {
    v2f a, b;
    a.x = xt[row * DIM + kk + koff];
    a.y = xt[row * DIM + kk + koff + 1];
    b.x = W[(kk + koff) * HU + colg];
    b.y = W[(kk + koff + 1) * HU + colg];
    c = __builtin_amdgcn_wmma_f32_16x16x4_f32(false, a, false, b, (short)0, c,
                                              false, false);
  }
  #pragma unroll
  for (int j = 0; j < 8; ++j) {
    const int r = (lane >> 4) * 8 + j;
    out[(n0 + r) * HU + colg] = c[j];
  }
}

// ---------------------------------------------------------------------------
// 2) Fused edge projection + attention scores + segment max.
//    eh = EF[64-edge tile] @ We + be computed by WMMA straight into LDS
//    (never hits HBM: saves ~820MB of round-trip traffic), then
//    s[e,h] = sum_u (q[src]+eh)[h,u] * k[dst][h,u], atomicMax per (src,h).
//    grid E/64, block 128 (4 waves x 16-edge WMMA tiles)
// ---------------------------------------------------------------------------
__global__ __launch_bounds__(128) void edge_scores(
    const float* __restrict__ EF, const float* __restrict__ We,
    const float* __restrict__ be,
    const int* __restrict__ src, const int* __restrict__ dst,
    const float* __restrict__ Q, const float* __restrict__ Km,
    float* __restrict__ S, unsigned* __restrict__ smaxu) {
  __shared__ float ef[64 * EDIM];  // 16KB staged edge features
  __shared__ float eh[64 * HU];    // 32KB projected edge features (LDS only)
  const int e0 = blockIdx.x * 64;

  for (int i = threadIdx.x; i < 64 * EDIM; i += 128)
    ef[i] = EF[e0 * EDIM + i];
  if (e0 + 64 < N_EDGES)  // speculative prefetch of next tile into caches
    __builtin_prefetch(&EF[(size_t)(e0 + 64) * EDIM + threadIdx.x * 4], 0, 1);
  __syncthreads();

  const int wave = threadIdx.x >> 5, lane = threadIdx.x & 31;
  const int ew0  = wave * 16;           // this wave's 16-edge sub-tile
  const int rowl = lane & 15;
  const int koff = (lane >> 4) * 2;

  #pragma unroll
  for (int t = 0; t < 8; ++t) {         // 8 column tiles of 16 (HU=128)
    const int col = t * 16 + rowl;
    v8f c;
    const float bb = be[col];
    #pragma unroll
    for (int j = 0; j < 8; ++j) c[j] = bb;
    #pragma unroll
    for (int kk = 0; kk < EDIM; kk += 4) {
      v2f a, b;
      a.x = ef[(ew0 + rowl) * EDIM + kk + koff];
      a.y = ef[(ew0 + rowl) * EDIM + kk + koff + 1];
      b.x = We[(kk + koff) * HU + col];
      b.y = We[(kk + koff + 1) * HU + col];
      c = __builtin_amdgcn_wmma_f32_16x16x4_f32(false, a, false, b, (short)0, c,
                                                false, false);
    }
    #pragma unroll
    for (int j = 0; j < 8; ++j)
      eh[(ew0 + (lane >> 4) * 8 + j) * HU + col] = c[j];
  }
  __syncthreads();

  // scores: 64 edges x 4 heads = 256 pairs over 128 threads
  for (int p = threadIdx.x; p < 256; p += 128) {
    const int el = p >> 2, h = p & 3;
    const int e = e0 + el;
    const int s = src[e], d = dst[e];
    const float4* qp = reinterpret_cast<const float4*>(Q  + (size_t)s * HU + h * 32);
    const float4* kp = reinterpret_cast<const float4*>(Km + (size_t)d * HU + h * 32);
    const float4* ep = reinterpret_cast<const float4*>(eh + el * HU + h * 32);
    float acc = 0.0f;
    #pragma unroll
    for (int u = 0; u < 8; ++u) {
      const float4 qv = qp[u], kv = kp[u], ev = ep[u];
      acc += (qv.x + ev.x) * kv.x + (qv.y + ev.y) * kv.y +
             (qv.z + ev.z) * kv.z + (qv.w + ev.w) * kv.w;
    }
    S[e * 4 + h] = acc;
    atomicMax(&smaxu[s * 4 + h], f32_order(acc));
  }
}

// ---------------------------------------------------------------------------
// 3) Unnormalized softmax accumulation:
//    num[src,:] += exp(s - max) * v[dst,:]   (coalesced f32 atomics, L2-resident)
//    den[src,h] += exp(s - max)
//    grid E/64, block 256 (one wave per 8 edges; lane == column within head)
// ---------------------------------------------------------------------------
__global__ __launch_bounds__(256) void softmax_accum(
    const int* __restrict__ src, const int* __restrict__ dst,
    const float* __restrict__ S, const unsigned* __restrict__ smaxu,
    const float* __restrict__ V,
    float* __restrict__ num, float* __restrict__ den) {
  const int wave = threadIdx.x >> 5, lane = threadIdx.x & 31;
  const int ebase = blockIdx.x * 64 + wave * 8;
  for (int i = 0; i < 8; ++i) {
    const int e = ebase + i;
    const int s = src[e], d = dst[e];
    #pragma unroll
    for (int h = 0; h < HEADS; ++h) {
      const float mx = f32_unorder(smaxu[s * 4 + h]);
      const float ex = __expf(S[e * 4 + h] - mx);
      const int col = h * 32 + lane;
      atomicAdd(&num[(size_t)s * HU + col], ex * V[(size_t)d * HU + col]);
      if (lane == 0) atomicAdd(&den[s * 4 + h], ex);
    }
  }
}

// ---------------------------------------------------------------------------
// 4) Output projection: relu((num/den) @ Wo + bo), fp32 WMMA
//    grid N/16, block 64 (2 waves = 2 column tiles of 16, UOUT=32)
// ---------------------------------------------------------------------------
__global__ __launch_bounds__(64) void out_proj(
    const float* __restrict__ num, const float* __restrict__ den,
    const float* __restrict__ Wo, const float* __restrict__ bo,
    float* __restrict__ out) {
  const int n0 = blockIdx.x * 16;
  const int wave = threadIdx.x >> 5, lane = threadIdx.x & 31;
  const int colg = wave * 16 + (lane & 15);
  const int rowg = n0 + (lane & 15);
  const int koff = (lane >> 4) * 2;

  float invd[4];
  #pragma unroll
  for (int h = 0; h < HEADS; ++h) {
    const float dv = den[rowg * 4 + h];
    invd[h] = dv > 0.0f ? 1.0f / dv : 0.0f;  // empty segments -> 0 (matches ref)
  }

  v8f c = {0, 0, 0, 0, 0, 0, 0, 0};
  #pragma unroll
  for (int kk = 0; kk < HU; kk += 4) {       // head index static under unroll
    const int h = kk >> 5;
    v2f a, b;
    a.x = num[(size_t)rowg * HU + kk + koff] * invd[h];
    a.y = num[(size_t)rowg * HU + kk + koff + 1] * invd[h];
    b.x = Wo[(kk + koff) * UOUT + colg];
    b.y = Wo[(kk + koff + 1) * UOUT + colg];
    c = __builtin_amdgcn_wmma_f32_16x16x4_f32(false, a, false, b, (short)0, c,
                                              false, false);
  }
  const float bb = bo[colg];
  #pragma unroll
  for (int j = 0; j < 8; ++j) {
    const int r = n0 + (lane >> 4) * 8 + j;
    const float v = c[j] + bb;
    out[r * UOUT + colg] = v > 0.0f ? v : 0.0f;
  }
}

// ---------------------------------------------------------------------------
extern "C" void kernel_launch(void* const* d_in, const int* in_sizes, int n_in,
                              void* d_out, int out_size, void* d_ws, size_t ws_size,
                              hipStream_t stream) {
  const float* X  = (const float*)d_in[0];
  const int*   ei = (const int*)  d_in[1];
  const float* EF = (const float*)d_in[2];
  const float* Wq = (const float*)d_in[3];
  const float* bq = (const float*)d_in[4];
  const float* Wk = (const float*)d_in[5];
  const float* bk = (const float*)d_in[6];
  const float* Wv = (const float*)d_in[7];
  const float* bv = (const float*)d_in[8];
  const float* We = (const float*)d_in[9];
  const float* be = (const float*)d_in[10];
  const float* Wo = (const float*)d_in[11];
  const float* bo = (const float*)d_in[12];
  float* out = (float*)d_out;
  const int* src = ei;             // edge_index[0]
  const int* dst = ei + N_EDGES;   // edge_index[1]

  char* ws = (char*)d_ws;
  float*    Q     = (float*)ws;    ws += (size_t)N_NODES * HU * 4;
  float*    K     = (float*)ws;    ws += (size_t)N_NODES * HU * 4;
  float*    V     = (float*)ws;    ws += (size_t)N_NODES * HU * 4;
  float*    S     = (float*)ws;    ws += (size_t)N_EDGES * HEADS * 4;
  float*    num   = (float*)ws;    ws += (size_t)N_NODES * HU * 4;
  float*    den   = (float*)ws;    ws += (size_t)N_NODES * HEADS * 4;
  unsigned* smaxu = (unsigned*)ws; ws += (size_t)N_NODES * HEADS * 4;

  init_ws<<<(N_NODES * HU + 255) / 256, 256, 0, stream>>>(num, den, smaxu);
  proj_qkv<<<dim3(N_NODES / 16, 3), 256, 0, stream>>>(X, Wq, bq, Wk, bk, Wv, bv,
                                                      Q, K, V);
  edge_scores<<<N_EDGES / 64, 128, 0, stream>>>(EF, We, be, src, dst, Q, K, S,
                                                smaxu);
  softmax_accum<<<N_EDGES / 64, 256, 0, stream>>>(src, dst, S, smaxu, V, num,
                                                  den);
  out_proj<<<N_NODES / 16, 64, 0, stream>>>(num, den, Wo, bo, out);
}